// BoundaryRefiner_558345749226
// MI455X (gfx1250) — compile-verified
//
#include <hip/hip_runtime.h>
#include <hip/hip_bf16.h>
#include <math.h>

typedef float v2f __attribute__((ext_vector_type(2)));
typedef float v8f __attribute__((ext_vector_type(8)));

#define BN_EPS 1e-5f
#define PI4 0.78539816339744830962f

// D(16x16) = A(16x4) * B(4x16) + C, fp32 WMMA (CDNA5)
static __device__ __forceinline__ v8f wmma4(v2f a, v2f b, v8f c) {
  return __builtin_amdgcn_wmma_f32_16x16x4_f32(
      /*neg_a=*/false, a, /*neg_b=*/false, b,
      /*c_mod=*/(short)0, c, /*reuse_a=*/false, /*reuse_b=*/false);
}

// ---------------------------------------------------------------- stage 1
// pooled[b*C+c] = mean over H*W of x[b,c,:,:]
__global__ void pool_kernel(const float* __restrict__ x, float* __restrict__ pooled) {
  const int bc = blockIdx.x;          // 0..511
  const int tid = threadIdx.x;        // 256 threads
  float s = 0.f;
  const float* p = x + (size_t)bc * 16384;
  for (int t = tid; t < 16384; t += 256) s += p[t];
  __shared__ float red[256];
  red[tid] = s;
  __syncthreads();
  for (int off = 128; off > 0; off >>= 1) {
    if (tid < off) red[tid] += red[tid + off];
    __syncthreads();
  }
  if (tid == 0) pooled[bc] = red[0] * (1.f / 16384.f);
}

// ---------------------------------------------------------------- stage 2
// per (branch i, batch b): angle MLP -> rotated kernels; meta MLP -> modulation
__global__ void head_kernel(const float* __restrict__ pooled,
                            const float* __restrict__ bk,
                            const float* __restrict__ aw1, const float* __restrict__ ab1,
                            const float* __restrict__ aw2, const float* __restrict__ ab2,
                            const float* __restrict__ mw1, const float* __restrict__ mb1,
                            const float* __restrict__ mw2, const float* __restrict__ mb2,
                            float* __restrict__ mod_ws, float* __restrict__ rot_ws) {
  const int i = blockIdx.x >> 3;   // branch
  const int b = blockIdx.x & 7;    // batch
  const int tid = threadIdx.x;     // 64 threads

  __shared__ float p[64], h1[16], mh[16], ang[8];
  p[tid] = pooled[b * 64 + tid];
  __syncthreads();

  if (tid < 16) {
    float s = ab1[i * 16 + tid];
    for (int c = 0; c < 64; ++c) s += aw1[(i * 16 + tid) * 64 + c] * p[c];
    h1[tid] = fmaxf(s, 0.f);
  } else if (tid < 32) {
    const int j = tid - 16;
    float s = mb1[i * 16 + j];
    for (int c = 0; c < 64; ++c) s += mw1[(i * 16 + j) * 64 + c] * p[c];
    mh[j] = fmaxf(s, 0.f);
  }
  __syncthreads();

  if (tid < 8) {
    float s = ab2[i * 8 + tid];
    for (int j = 0; j < 16; ++j) s += aw2[(i * 8 + tid) * 16 + j] * h1[j];
    ang[tid] = tanhf(s) * PI4;
  }
  {
    float s = mb2[i * 64 + tid];
    for (int j = 0; j < 16; ++j) s += mw2[(i * 64 + tid) * 16 + j] * mh[j];
    mod_ws[(i * 8 + b) * 64 + tid] = 1.f / (1.f + expf(-s));
  }
  __syncthreads();

  // rotate kernels: 8 groups * 8 * 8 * 9 = 4608 elements
  for (int e = tid; e < 4608; e += 64) {
    const int k9 = e % 9;
    const int ci = (e / 9) & 7;
    const int co = (e / 72) & 7;
    const int g  = e / 576;
    const int r  = k9 / 3, s2 = k9 % 3;
    const float a = ang[g];
    const float ct = cosf(a), st = sinf(a);
    const float yy = (float)r - 1.f, xx = (float)s2 - 1.f;
    const float xr = xx * ct + yy * st + 1.f;
    const float yr = -xx * st + yy * ct + 1.f;
    float x0 = fminf(fmaxf(floorf(xr), 0.f), 1.f);
    float y0 = fminf(fmaxf(floorf(yr), 0.f), 1.f);
    const float x1 = x0 + 1.f, y1 = y0 + 1.f;
    const float w00 = (x1 - xr) * (y1 - yr);
    const float w01 = (x1 - xr) * (yr - y0);
    const float w10 = (xr - x0) * (y1 - yr);
    const float w11 = (xr - x0) * (yr - y0);
    const int x0i = (int)x0, y0i = (int)y0;
    const float* kb = bk + ((((size_t)(i * 8 + g) * 8 + co) * 8 + ci) * 9);
    const float v = w00 * kb[y0i * 3 + x0i]     + w01 * kb[(y0i + 1) * 3 + x0i] +
                    w10 * kb[y0i * 3 + x0i + 1] + w11 * kb[(y0i + 1) * 3 + x0i + 1];
    rot_ws[(size_t)(i * 8 + b) * 4608 + e] = v;
  }
}

// ---------------------------------------------------------------- stage 3
// grouped 3x3 conv (+mod scale) -> feats[branch][b][c][H*W]
__global__ void __launch_bounds__(256)
conv_kernel(const float* __restrict__ x, const float* __restrict__ rot,
            const float* __restrict__ mod, float* __restrict__ feats) {
  const int tile = blockIdx.x;   // 0..63 (8x8 tiles of 16x16)
  const int g    = blockIdx.y;   // 0..7
  const int ib   = blockIdx.z;   // branch*8 + batch
  const int i = ib >> 3, b = ib & 7;
  const int tid = threadIdx.x;
  const int ty = (tile >> 3) * 16, tx = (tile & 7) * 16;

  __shared__ float xs[8 * 18 * 18];   // 8 in-channels, 16x16 tile + halo
  __shared__ float wl[8 * 8 * 9];
  __shared__ float mv[8];

  for (int t = tid; t < 2592; t += 256) {
    const int ci = t / 324, rem = t % 324;
    const int yy = rem / 18, xx = rem % 18;
    const int gy = ty + yy - 1, gx = tx + xx - 1;
    float v = 0.f;
    if (gy >= 0 && gy < 128 && gx >= 0 && gx < 128)
      v = x[((size_t)(b * 64 + g * 8 + ci)) * 16384 + gy * 128 + gx];
    xs[t] = v;
  }
  for (int t = tid; t < 576; t += 256)
    wl[t] = rot[(size_t)((i * 8 + b) * 8 + g) * 576 + t];
  if (tid < 8) mv[tid] = mod[(i * 8 + b) * 64 + g * 8 + tid];
  __syncthreads();

  const int py = tid >> 4, px = tid & 15;
  #pragma unroll
  for (int co = 0; co < 8; ++co) {
    float acc = 0.f;
    #pragma unroll
    for (int ci = 0; ci < 8; ++ci) {
      const float* xr = xs + ci * 324 + py * 18 + px;
      const float* wr = wl + (co * 8 + ci) * 9;
      #pragma unroll
      for (int ky = 0; ky < 3; ++ky)
        #pragma unroll
        for (int kx = 0; kx < 3; ++kx)
          acc = fmaf(xr[ky * 18 + kx], wr[ky * 3 + kx], acc);
    }
    feats[(((size_t)i * 8 + b) * 64 + g * 8 + co) * 16384 + (ty + py) * 128 + tx + px]
        = acc * mv[co];
  }
}

// ---------------------------------------------------------------- stage 4
// Per-pixel channel MLPs on WMMA: dir GEMM (K=256,M=64) -> BN/ReLU -> attention
// softmax -> fused*roi -> edge GEMM1 (K=64,M=32) -> BN/ReLU -> edge GEMM2
// (K=32,M=64) -> sigmoid -> out = x*(1+e). One wave handles 16 pixels.
#define YB 1088  // 64 rows * 17 (padded)
__global__ void __launch_bounds__(256)
fuse_kernel(const float* __restrict__ feats, const float* __restrict__ x,
            const float* __restrict__ roi,
            const float* __restrict__ dw1,
            const float* __restrict__ dbnw, const float* __restrict__ dbnb,
            const float* __restrict__ dbnm, const float* __restrict__ dbnv,
            const float* __restrict__ dw2, const float* __restrict__ db2,
            const float* __restrict__ ew1,
            const float* __restrict__ ebnw, const float* __restrict__ ebnb,
            const float* __restrict__ ebnm, const float* __restrict__ ebnv,
            const float* __restrict__ ew2, const float* __restrict__ eb2g,
            float* __restrict__ out) {
  const int tid  = threadIdx.x;
  const int wave = tid >> 5;
  const int lane = tid & 31;
  const int pxl  = lane & 15;
  const int b    = blockIdx.y;
  const int px0  = blockIdx.x * 128 + wave * 16;
  const size_t pix = (size_t)(px0 + pxl);

  __shared__ float w1s[64 * 256];   // dir_w1
  __shared__ float e1w[32 * 64];    // edge_w1
  __shared__ float e2w[64 * 32];    // edge_w2
  __shared__ float w2s[256], b2s[4], eb2s[64];
  __shared__ float dsc[64], dsh[64], esc[32], esh[32];
  __shared__ float ybuf[8 * YB];    // per-wave scratch: y -> fused*roi -> e1
  __shared__ float abuf[8 * 64];    // per-wave attention [4][16]

  for (int t = tid; t < 16384; t += 256) w1s[t] = dw1[t];
  for (int t = tid; t < 2048;  t += 256) e1w[t] = ew1[t];
  for (int t = tid; t < 2048;  t += 256) e2w[t] = ew2[t];
  if (tid < 256) w2s[tid] = dw2[tid];
  if (tid < 4)   b2s[tid] = db2[tid];
  if (tid < 64) {
    eb2s[tid] = eb2g[tid];
    const float sc = dbnw[tid] * rsqrtf(dbnv[tid] + BN_EPS);
    dsc[tid] = sc;
    dsh[tid] = dbnb[tid] - dbnm[tid] * sc;
  }
  if (tid < 32) {
    const float sc = ebnw[tid] * rsqrtf(ebnv[tid] + BN_EPS);
    esc[tid] = sc;
    esh[tid] = ebnb[tid] - ebnm[tid] * sc;
  }
  __syncthreads();

  const int  kcoff = (lane < 16) ? 0 : 2;  // K lane split (ISA A/B layout)
  const int  rb    = (lane < 16) ? 0 : 8;  // C/D M offset per half-wave
  float* yw = ybuf + wave * YB;

  // ---- dir GEMM: y[64][16] = dir_w1[64][256] @ featcat[256][16]
  v8f acc0 = {0,0,0,0,0,0,0,0}, acc1 = acc0, acc2 = acc0, acc3 = acc0;
  for (int kk = 0; kk < 64; ++kk) {
    const int cA = kk * 4 + kcoff;
    const int br = cA >> 6, ch = cA & 63;
    const float* fp = feats + (((size_t)br * 8 + b) * 64 + ch) * 16384 + pix;
    v2f bb; bb.x = fp[0]; bb.y = fp[16384];
    v2f aa;
    aa.x = w1s[(0  + pxl) * 256 + cA]; aa.y = w1s[(0  + pxl) * 256 + cA + 1];
    acc0 = wmma4(aa, bb, acc0);
    aa.x = w1s[(16 + pxl) * 256 + cA]; aa.y = w1s[(16 + pxl) * 256 + cA + 1];
    acc1 = wmma4(aa, bb, acc1);
    aa.x = w1s[(32 + pxl) * 256 + cA]; aa.y = w1s[(32 + pxl) * 256 + cA + 1];
    acc2 = wmma4(aa, bb, acc2);
    aa.x = w1s[(48 + pxl) * 256 + cA]; aa.y = w1s[(48 + pxl) * 256 + cA + 1];
    acc3 = wmma4(aa, bb, acc3);
  }
  #pragma unroll
  for (int r = 0; r < 8; ++r) {
    int o;
    o = 0  + rb + r; yw[o * 17 + pxl] = fmaxf(acc0[r] * dsc[o] + dsh[o], 0.f);
    o = 16 + rb + r; yw[o * 17 + pxl] = fmaxf(acc1[r] * dsc[o] + dsh[o], 0.f);
    o = 32 + rb + r; yw[o * 17 + pxl] = fmaxf(acc2[r] * dsc[o] + dsh[o], 0.f);
    o = 48 + rb + r; yw[o * 17 + pxl] = fmaxf(acc3[r] * dsc[o] + dsh[o], 0.f);
  }
  __syncthreads();

  // ---- attention logits + softmax (one lane per pixel)
  if (lane < 16) {
    float lg[4];
    #pragma unroll
    for (int q = 0; q < 4; ++q) {
      float s = b2s[q];
      for (int o = 0; o < 64; ++o) s += w2s[q * 64 + o] * yw[o * 17 + lane];
      lg[q] = s;
    }
    const float mx = fmaxf(fmaxf(lg[0], lg[1]), fmaxf(lg[2], lg[3]));
    const float e0 = expf(lg[0] - mx), e1 = expf(lg[1] - mx);
    const float e2 = expf(lg[2] - mx), e3 = expf(lg[3] - mx);
    const float inv = 1.f / (e0 + e1 + e2 + e3);
    abuf[wave * 64 + 0 * 16 + lane] = e0 * inv;
    abuf[wave * 64 + 1 * 16 + lane] = e1 * inv;
    abuf[wave * 64 + 2 * 16 + lane] = e2 * inv;
    abuf[wave * 64 + 3 * 16 + lane] = e3 * inv;
  }
  __syncthreads();

  // ---- roi bilinear (align_corners) for this lane's pixel
  float rv;
  {
    const int gp = px0 + pxl;
    const int py = gp >> 7, pxx = gp & 127;
    const float S = 63.f / 127.f;
    const float ys = py * S, xs = pxx * S;
    float yf = fminf(floorf(ys), 62.f), xf = fminf(floorf(xs), 62.f);
    const int y0i = (int)yf, x0i = (int)xf;
    const float wy = ys - yf, wx = xs - xf;
    const float* rm = roi + (size_t)b * 4096;
    const float v00 = rm[y0i * 64 + x0i],       v01 = rm[y0i * 64 + x0i + 1];
    const float v10 = rm[(y0i + 1) * 64 + x0i], v11 = rm[(y0i + 1) * 64 + x0i + 1];
    rv = v00 * (1.f - wy) * (1.f - wx) + v01 * (1.f - wy) * wx +
         v10 * wy * (1.f - wx)         + v11 * wy * wx;
  }

  // ---- fused = sum_i feats_i * attn_i, times roi (reuse ybuf)
  {
    const float a0 = abuf[wave * 64 + 0 * 16 + pxl];
    const float a1 = abuf[wave * 64 + 1 * 16 + pxl];
    const float a2 = abuf[wave * 64 + 2 * 16 + pxl];
    const float a3 = abuf[wave * 64 + 3 * 16 + pxl];
    const size_t FSZ = (size_t)8 * 64 * 16384;
    const int cb = (lane < 16) ? 0 : 32;
    for (int cc = 0; cc < 32; ++cc) {
      const int c = cb + cc;
      const float* f0 = feats + ((size_t)b * 64 + c) * 16384 + pix;
      const float fv = f0[0] * a0 + f0[FSZ] * a1 + f0[2 * FSZ] * a2 + f0[3 * FSZ] * a3;
      yw[c * 17 + pxl] = fv * rv;
    }
  }
  __syncthreads();

  // ---- edge GEMM1: e1[32][16] = edge_w1[32][64] @ fr[64][16]
  v8f ea0 = {0,0,0,0,0,0,0,0}, ea1 = ea0;
  #pragma unroll
  for (int kk = 0; kk < 16; ++kk) {
    const int cA = kk * 4 + kcoff;
    v2f bb; bb.x = yw[cA * 17 + pxl]; bb.y = yw[(cA + 1) * 17 + pxl];
    v2f aa;
    aa.x = e1w[(0  + pxl) * 64 + cA]; aa.y = e1w[(0  + pxl) * 64 + cA + 1];
    ea0 = wmma4(aa, bb, ea0);
    aa.x = e1w[(16 + pxl) * 64 + cA]; aa.y = e1w[(16 + pxl) * 64 + cA + 1];
    ea1 = wmma4(aa, bb, ea1);
  }
  __syncthreads();
  #pragma unroll
  for (int r = 0; r < 8; ++r) {
    int o;
    o = 0  + rb + r; yw[o * 17 + pxl] = fmaxf(ea0[r] * esc[o] + esh[o], 0.f);
    o = 16 + rb + r; yw[o * 17 + pxl] = fmaxf(ea1[r] * esc[o] + esh[o], 0.f);
  }
  __syncthreads();

  // ---- edge GEMM2: e2[64][16] = edge_w2[64][32] @ e1[32][16]
  v8f oa0 = {0,0,0,0,0,0,0,0}, oa1 = oa0, oa2 = oa0, oa3 = oa0;
  #pragma unroll
  for (int kk = 0; kk < 8; ++kk) {
    const int cA = kk * 4 + kcoff;
    v2f bb; bb.x = yw[cA * 17 + pxl]; bb.y = yw[(cA + 1) * 17 + pxl];
    v2f aa;
    aa.x = e2w[(0  + pxl) * 32 + cA]; aa.y = e2w[(0  + pxl) * 32 + cA + 1];
    oa0 = wmma4(aa, bb, oa0);
    aa.x = e2w[(16 + pxl) * 32 + cA]; aa.y = e2w[(16 + pxl) * 32 + cA + 1];
    oa1 = wmma4(aa, bb, oa1);
    aa.x = e2w[(32 + pxl) * 32 + cA]; aa.y = e2w[(32 + pxl) * 32 + cA + 1];
    oa2 = wmma4(aa, bb, oa2);
    aa.x = e2w[(48 + pxl) * 32 + cA]; aa.y = e2w[(48 + pxl) * 32 + cA + 1];
    oa3 = wmma4(aa, bb, oa3);
  }

  // ---- sigmoid gate and residual output
  #pragma unroll
  for (int r = 0; r < 8; ++r) {
    #pragma unroll
    for (int mt = 0; mt < 4; ++mt) {
      const int o = mt * 16 + rb + r;
      const float v = (mt == 0 ? oa0[r] : mt == 1 ? oa1[r] : mt == 2 ? oa2[r] : oa3[r])
                      + eb2s[o];
      const float sg = 1.f / (1.f + expf(-v));
      const size_t oi = ((size_t)b * 64 + o) * 16384 + pix;
      out[oi] = x[oi] * (1.f + sg);
    }
  }
}

// ---------------------------------------------------------------- launch
extern "C" void kernel_launch(void* const* d_in, const int* in_sizes, int n_in,
                              void* d_out, int out_size, void* d_ws, size_t ws_size,
                              hipStream_t stream) {
  (void)in_sizes; (void)n_in; (void)out_size; (void)ws_size;
  const float* x    = (const float*)d_in[0];
  const float* roi  = (const float*)d_in[1];
  const float* bk   = (const float*)d_in[2];
  const float* aw1  = (const float*)d_in[3];
  const float* ab1  = (const float*)d_in[4];
  const float* aw2  = (const float*)d_in[5];
  const float* ab2  = (const float*)d_in[6];
  const float* mw1  = (const float*)d_in[7];
  const float* mb1  = (const float*)d_in[8];
  const float* mw2  = (const float*)d_in[9];
  const float* mb2  = (const float*)d_in[10];
  const float* dw1  = (const float*)d_in[11];
  const float* dbnw = (const float*)d_in[12];
  const float* dbnb = (const float*)d_in[13];
  const float* dbnm = (const float*)d_in[14];
  const float* dbnv = (const float*)d_in[15];
  const float* dw2  = (const float*)d_in[16];
  const float* db2  = (const float*)d_in[17];
  const float* ew1  = (const float*)d_in[18];
  const float* ebnw = (const float*)d_in[19];
  const float* ebnb = (const float*)d_in[20];
  const float* ebnm = (const float*)d_in[21];
  const float* ebnv = (const float*)d_in[22];
  const float* ew2  = (const float*)d_in[23];
  const float* eb2  = (const float*)d_in[24];

  float* ws     = (float*)d_ws;
  float* pooled = ws;              // 512
  float* mod    = ws + 768;        // 2048
  float* rot    = ws + 2816;       // 147456
  float* feats  = ws + 150272;     // 4*8*64*16384 = 33,554,432 floats (~128 MB, L2-resident)

  pool_kernel<<<512, 256, 0, stream>>>(x, pooled);
  head_kernel<<<32, 64, 0, stream>>>(pooled, bk, aw1, ab1, aw2, ab2,
                                     mw1, mb1, mw2, mb2, mod, rot);
  conv_kernel<<<dim3(64, 8, 32), 256, 0, stream>>>(x, rot, mod, feats);
  fuse_kernel<<<dim3(128, 8), 256, 0, stream>>>(feats, x, roi, dw1,
                                                dbnw, dbnb, dbnm, dbnv, dw2, db2,
                                                ew1, ebnw, ebnb, ebnm, ebnv, ew2, eb2,
                                                (float*)d_out);
}